// ChannelFeatures_2843268350263
// MI455X (gfx1250) — compile-verified
//
#include <hip/hip_runtime.h>
#include <hip/hip_bf16.h>
#include <math.h>

typedef float v2f __attribute__((ext_vector_type(2)));
typedef float v4f __attribute__((ext_vector_type(4)));
typedef float v8f __attribute__((ext_vector_type(8)));

#define DIMS 64
#define CONV_NUM 7
#define BATCH 16
#define HW 65536                      // 256*256
#define NCHUNK 64                     // chunks per batch image
#define PIX_PER_CHUNK (HW / NCHUNK)   // 1024 pixels
#define PPG (PIX_PER_CHUNK / 16)      // 64 pixels per thread-group

// ---------------------------------------------------------------------------
// Kernel 1: per-chunk max/sum partial reduction over spatial dims.
// features layout: (B, H*W, C) channels-last, C = 64.
// Block = 256 threads: 16 channel-quads (float4) x 16 pixel groups.
// ---------------------------------------------------------------------------
__global__ __launch_bounds__(256) void cf_reduce_kernel(const float* __restrict__ feat,
                                                        float* __restrict__ pmax,
                                                        float* __restrict__ psum) {
  const int b     = blockIdx.x / NCHUNK;
  const int chunk = blockIdx.x % NCHUNK;
  const int t  = threadIdx.x;
  const int g  = t >> 4;          // 0..15 pixel group
  const int c4 = (t & 15) << 2;   // channel quad start

  const v4f* base = (const v4f*)(feat +
      ((size_t)b * HW + (size_t)chunk * PIX_PER_CHUNK + (size_t)g * PPG) * DIMS + c4);

  const float ninf = -__builtin_inff();
  float mx0 = ninf, mx1 = ninf, mx2 = ninf, mx3 = ninf;
  float s0 = 0.f, s1 = 0.f, s2 = 0.f, s3 = 0.f;
#pragma unroll 8
  for (int i = 0; i < PPG; ++i) {
    v4f v = base[(size_t)i * (DIMS / 4)];
    mx0 = fmaxf(mx0, v.x); s0 += v.x;
    mx1 = fmaxf(mx1, v.y); s1 += v.y;
    mx2 = fmaxf(mx2, v.z); s2 += v.z;
    mx3 = fmaxf(mx3, v.w); s3 += v.w;
  }

  __shared__ float smax[16][DIMS];
  __shared__ float ssum[16][DIMS];
  smax[g][c4 + 0] = mx0; smax[g][c4 + 1] = mx1; smax[g][c4 + 2] = mx2; smax[g][c4 + 3] = mx3;
  ssum[g][c4 + 0] = s0;  ssum[g][c4 + 1] = s1;  ssum[g][c4 + 2] = s2;  ssum[g][c4 + 3] = s3;
  __syncthreads();

  if (t < DIMS) {
    float m = smax[0][t], s = ssum[0][t];
#pragma unroll
    for (int gg = 1; gg < 16; ++gg) { m = fmaxf(m, smax[gg][t]); s += ssum[gg][t]; }
    const size_t o = ((size_t)b * NCHUNK + chunk) * DIMS + t;
    pmax[o] = m;
    psum[o] = s;
  }
}

// ---------------------------------------------------------------------------
// Kernel 2: finish reduction, then wave 0 runs the 7 residual blocks with
// V_WMMA_F32_16X16X4_F32 (exact fp32), then sigmoid -> scores (16x64).
//
// out = x(16x64) @ W(64x64)^T per layer:
//   out[M][j0+N] = sum_K x[M][K] * W[j0+N][K]
// A fragment (16x4):  lane L: M = L&15, A.{x,y} = x[M][k0+2*(L>>4) + {0,1}]
// B fragment (4x16):  lane L: N = L&15, B.{x,y} = W[j0+N][k0+2*(L>>4) + {0,1}]
// C/D (16x16):        lane L, vgpr v: M = v + 8*(L>>4), N = L&15
// ---------------------------------------------------------------------------
__global__ __launch_bounds__(1024) void cf_mlp_kernel(const float* __restrict__ pmax,
                                                      const float* __restrict__ psum,
                                                      const float* __restrict__ W1,
                                                      const float* __restrict__ b1,
                                                      const float* __restrict__ a1,
                                                      float* __restrict__ scores,
                                                      float inv_count) {
  __shared__ float xbuf[2][2][BATCH * DIMS];  // [branch max/avg][ping-pong][16*64]

  const int tid = threadIdx.x;
  const int b = tid >> 6;
  const int c = tid & 63;

  // finish the (b,c) reduction over NCHUNK partials
  float vmax = -__builtin_inff();
  float vsum = 0.f;
  const float* pm = pmax + (size_t)b * NCHUNK * DIMS + c;
  const float* ps = psum + (size_t)b * NCHUNK * DIMS + c;
#pragma unroll 8
  for (int ch = 0; ch < NCHUNK; ++ch) {
    vmax = fmaxf(vmax, pm[(size_t)ch * DIMS]);
    vsum += ps[(size_t)ch * DIMS];
  }
  xbuf[0][0][b * DIMS + c] = vmax;
  xbuf[1][0][b * DIMS + c] = vsum * inv_count;
  __syncthreads();

  if (tid < 32) {                 // wave 0 only: EXEC all ones, as WMMA requires
    const int lane = tid;
    const int h = lane >> 4;      // lane half
    const int n = lane & 15;      // A row / B column / D column

    for (int k = 0; k < CONV_NUM; ++k) {
      const float* Wk = W1 + (size_t)k * DIMS * DIMS;
      const float alpha = a1[k];
      const int cur = k & 1, nxt = cur ^ 1;
      float* xm_in  = &xbuf[0][cur][0];
      float* xm_out = &xbuf[0][nxt][0];
      float* xa_in  = &xbuf[1][cur][0];
      float* xa_out = &xbuf[1][nxt][0];

      for (int j0 = 0; j0 < DIMS; j0 += 16) {
        const float bias = b1[k * DIMS + j0 + n];
        v8f accm, acca;
#pragma unroll
        for (int v = 0; v < 8; ++v) { accm[v] = bias; acca[v] = bias; }

#pragma unroll
        for (int k0 = 0; k0 < DIMS; k0 += 4) {
          const int kk = k0 + 2 * h;
          v2f bw; bw.x = Wk[(j0 + n) * DIMS + kk]; bw.y = Wk[(j0 + n) * DIMS + kk + 1];
          v2f am; am.x = xm_in[n * DIMS + kk];     am.y = xm_in[n * DIMS + kk + 1];
          v2f aa; aa.x = xa_in[n * DIMS + kk];     aa.y = xa_in[n * DIMS + kk + 1];
          accm = __builtin_amdgcn_wmma_f32_16x16x4_f32(false, am, false, bw,
                                                       (short)0, accm, false, false);
          acca = __builtin_amdgcn_wmma_f32_16x16x4_f32(false, aa, false, bw,
                                                       (short)0, acca, false, false);
        }

        // PReLU + residual; single-wave LDS ops are in order (ping-pong buffers)
#pragma unroll
        for (int v = 0; v < 8; ++v) {
          const int m = v + 8 * h;
          float ym = accm[v]; ym = (ym >= 0.f) ? ym : alpha * ym;
          float ya = acca[v]; ya = (ya >= 0.f) ? ya : alpha * ya;
          xm_out[m * DIMS + j0 + n] = ym + xm_in[m * DIMS + j0 + n];
          xa_out[m * DIMS + j0 + n] = ya + xa_in[m * DIMS + j0 + n];
        }
      }
    }

    const float* xm = &xbuf[0][CONV_NUM & 1][0];
    const float* xa = &xbuf[1][CONV_NUM & 1][0];
    for (int idx = lane; idx < BATCH * DIMS; idx += 32) {
      const float s = xm[idx] + xa[idx];
      scores[idx] = 1.f / (1.f + expf(-s));
    }
  }
}

// ---------------------------------------------------------------------------
// Kernel 3: out = features * scores[b, c] broadcast. float4 vectorized,
// nontemporal on the 512 MB streaming traffic (no reuse).
// ---------------------------------------------------------------------------
__global__ __launch_bounds__(256) void cf_scale_kernel(const float* __restrict__ feat,
                                                       const float* __restrict__ scores,
                                                       float* __restrict__ out) {
  const size_t i = (size_t)blockIdx.x * 256 + threadIdx.x;  // float4 index
  v4f f = __builtin_nontemporal_load((const v4f*)feat + i);
  const size_t b = i >> 20;       // HW*DIMS/4 = 1,048,576 float4 per batch
  const size_t q = i & 15;        // channel quad (DIMS/4 = 16)
  const v4f s = *((const v4f*)scores + (b << 4) + q);
  v4f o;
  o.x = f.x * s.x; o.y = f.y * s.y; o.z = f.z * s.z; o.w = f.w * s.w;
  __builtin_nontemporal_store(o, (v4f*)out + i);
}

extern "C" void kernel_launch(void* const* d_in, const int* in_sizes, int n_in,
                              void* d_out, int out_size, void* d_ws, size_t ws_size,
                              hipStream_t stream) {
  const float* feat = (const float*)d_in[0];   // (16, 256, 256, 64) fp32
  const float* W1   = (const float*)d_in[1];   // (7, 64, 64)
  const float* b1   = (const float*)d_in[2];   // (7, 64)
  const float* a1   = (const float*)d_in[3];   // (7,)
  float* out = (float*)d_out;

  float* pmax   = (float*)d_ws;                     // 16*64*64 floats
  float* psum   = pmax + BATCH * NCHUNK * DIMS;     // 16*64*64 floats
  float* scores = psum + BATCH * NCHUNK * DIMS;     // 16*64 floats

  cf_reduce_kernel<<<BATCH * NCHUNK, 256, 0, stream>>>(feat, pmax, psum);
  cf_mlp_kernel<<<1, 1024, 0, stream>>>(pmax, psum, W1, b1, a1, scores, 1.0f / (float)HW);
  const size_t n4 = (size_t)BATCH * HW * DIMS / 4;  // 16,777,216 float4
  cf_scale_kernel<<<(unsigned)(n4 / 256), 256, 0, stream>>>(feat, scores, out);
}